// FocalLoss_59176059404476
// MI455X (gfx1250) — compile-verified
//
#include <hip/hip_runtime.h>
#include <stdint.h>

#define A_ANCH 120000
#define B_IMG  8
#define C_CLS  80
#define M_GT   32
#define STAGES 5           // chunks per block in the focal stream kernel (>=2)
#define CHUNK_FLOATS 1024  // 256 threads * 4 floats

// d_ws layout (floats):
//  [0..7]   cls_sum[8]
//  [8..15]  reg_sum[8]
//  [16..23] npos[8]   (as unsigned bits)
//  byte offset 256:  uint8 state[B*A]  (254=negative, 255=ignore, else positive label)

__global__ void init_ws_kernel(float* __restrict__ ws) {
  int t = threadIdx.x;
  if (t < 24) ws[t] = 0.0f;
}

__device__ __forceinline__ float wave_sum32(float v) {
  #pragma unroll
  for (int off = 16; off > 0; off >>= 1) v += __shfl_xor(v, off, 32);
  return v;
}

// ---------------- Kernel 1: per-anchor assignment + regression loss ----------------
__global__ __launch_bounds__(256) void assign_kernel(
    const float* __restrict__ anchors,      // [A,4]
    const float* __restrict__ annotations,  // [B,M,5]
    const float* __restrict__ regressions,  // [B,A,4]
    float* __restrict__ ws)
{
  __shared__ float ann[M_GT * 5];
  const int b   = blockIdx.y;
  const int tid = threadIdx.x;
  const int a   = blockIdx.x * 256 + tid;

  if (tid < M_GT * 5) ann[tid] = annotations[b * M_GT * 5 + tid];
  __syncthreads();

  bool    pos = false;
  float   rl  = 0.0f;
  uint8_t st  = 254;

  if (a < A_ANCH) {
    float4 an = ((const float4*)anchors)[a];
    float aw = an.z - an.x, ah = an.w - an.y;
    float areaA = aw * ah;
    float best = -1e30f; int bestj = 0;
    #pragma unroll
    for (int j = 0; j < M_GT; ++j) {
      float gx1 = ann[j*5+0], gy1 = ann[j*5+1];
      float gx2 = ann[j*5+2], gy2 = ann[j*5+3];
      float lbl = ann[j*5+4];
      float iw = fminf(an.z, gx2) - fmaxf(an.x, gx1);
      float ih = fminf(an.w, gy2) - fmaxf(an.y, gy1);
      float inter = fmaxf(iw, 0.0f) * fmaxf(ih, 0.0f);
      float ua = areaA + (gx2 - gx1) * (gy2 - gy1) - inter;
      float iou = inter / fmaxf(ua, 1e-8f);
      if (lbl == -1.0f) iou = -1.0f;            // padded gt boxes ignored
      if (iou > best) { best = iou; bestj = j; } // strict > == jnp.argmax tie-break
    }
    pos = (best >= 0.5f);
    if (pos)                 st = (uint8_t)(int)ann[bestj*5+4];
    else if (best < 0.4f)    st = 254;
    else                     st = 255;

    if (pos) {
      float gx1 = ann[bestj*5+0], gy1 = ann[bestj*5+1];
      float gx2 = ann[bestj*5+2], gy2 = ann[bestj*5+3];
      float gw = gx2 - gx1, gh = gy2 - gy1;
      float gcx = gx1 + 0.5f * gw, gcy = gy1 + 0.5f * gh;  // uses pre-clamp gw/gh (matches ref)
      gw = fmaxf(gw, 1.0f); gh = fmaxf(gh, 1.0f);
      float acx = an.x + 0.5f * aw, acy = an.y + 0.5f * ah;
      float t0 = (gcx - acx) / aw * 10.0f;   // /0.1
      float t1 = (gcy - acy) / ah * 10.0f;
      float t2 = __logf(gw / aw) * 5.0f;     // /0.2
      float t3 = __logf(gh / ah) * 5.0f;
      float4 r = ((const float4*)regressions)[(size_t)b * A_ANCH + a];
      float d, s = 0.0f;
      d = fabsf(t0 - r.x); s += (d <= 1.0f/9.0f) ? 4.5f*d*d : d - (0.5f/9.0f);
      d = fabsf(t1 - r.y); s += (d <= 1.0f/9.0f) ? 4.5f*d*d : d - (0.5f/9.0f);
      d = fabsf(t2 - r.z); s += (d <= 1.0f/9.0f) ? 4.5f*d*d : d - (0.5f/9.0f);
      d = fabsf(t3 - r.w); s += (d <= 1.0f/9.0f) ? 4.5f*d*d : d - (0.5f/9.0f);
      rl = s;
    }
    uint8_t* state = (uint8_t*)ws + 256;
    state[(size_t)b * A_ANCH + a] = st;
  }

  // wave32 reductions: num_pos via ballot, reg-loss via shuffle tree
  unsigned long long bal = __ballot(pos);
  unsigned np = (unsigned)__popcll(bal);
  float rsum = wave_sum32(rl);
  if ((tid & 31) == 0) {
    if (np)          atomicAdd(((unsigned*)ws) + 16 + b, np);
    if (rsum != 0.f) atomicAdd(ws + 8 + b, rsum);
  }
}

// ---------------- Kernel 2: focal BCE over the 307 MB classification stream ----------------
// Depth-2 async global->LDS DMA pipeline (gfx1250 GLOBAL_LOAD_ASYNC_TO_LDS_B128, ASYNCcnt).
// Per wave per stage: one async instruction (ASYNCcnt += 1), in-order completion,
// so "asynccnt <= 1" proves stage s has landed while stage s+1 is still in flight.
__global__ __launch_bounds__(256) void focal_kernel(
    const float* __restrict__ cls,   // [B, A, C]
    float* __restrict__ ws)
{
  __shared__ __align__(16) float sbuf[2 * CHUNK_FLOATS];  // 8 KB double buffer
  __shared__ float wsum[8];

  const int b   = blockIdx.y;
  const int bx  = blockIdx.x;
  const int tid = threadIdx.x;

  const unsigned long long gbase =
      (unsigned long long)(uintptr_t)(cls + (size_t)b * ((size_t)A_ANCH * C_CLS)
                                          + (size_t)bx * (STAGES * CHUNK_FLOATS));
  const uint8_t* __restrict__ state = (const uint8_t*)ws + 256 + (size_t)b * A_ANCH;

  auto issue = [&](int s) {
    // LDS byte address: low 32 bits of the generic pointer are the LDS offset.
    unsigned ldsa = (unsigned)(uintptr_t)(void*)&sbuf[(s & 1) * CHUNK_FLOATS + tid * 4];
    unsigned goff = (unsigned)(s * (CHUNK_FLOATS * 4) + tid * 16);
    asm volatile("global_load_async_to_lds_b128 %0, %1, %2 offset:0 th:TH_LOAD_NT"
                 :: "v"(ldsa), "v"(goff), "s"(gbase)
                 : "memory");
  };

  issue(0);
  issue(1);
  float lsum = 0.0f;

  #pragma unroll
  for (int s = 0; s < STAGES; ++s) {
    if (s < STAGES - 1) {
      asm volatile("s_wait_asynccnt 0x1" ::: "memory");  // stage s landed, s+1 in flight
    } else {
      asm volatile("s_wait_asynccnt 0x0" ::: "memory");  // last stage landed
    }
    float4 v = *(const float4*)&sbuf[(s & 1) * CHUNK_FLOATS + tid * 4];
    asm volatile("s_wait_dscnt 0x0" ::: "memory");       // LDS read drained before buffer reuse
    if (s + 2 < STAGES) issue(s + 2);                    // refill the buffer just drained

    int e  = bx * (STAGES * CHUNK_FLOATS) + s * CHUNK_FLOATS + tid * 4; // element in image
    unsigned stv = state[e / C_CLS];                     // C=80 divisible by 4: one anchor per float4
    int c0 = e % C_CLS;

    float x[4] = {v.x, v.y, v.z, v.w};
    #pragma unroll
    for (int k = 0; k < 4; ++k) {
      float p    = fminf(fmaxf(x[k], 1e-4f), 1.0f - 1e-4f);
      float negL = 0.75f * p * p * (-__logf(1.0f - p));            // target 0
      float posL = 0.25f * (1.0f - p) * (1.0f - p) * (-__logf(p)); // target 1
      float l = (stv == 255u) ? 0.0f
              : (stv == 254u) ? negL
              : (((unsigned)(c0 + k) == stv) ? posL : negL);
      lsum += l;
    }
  }

  lsum = wave_sum32(lsum);
  if ((tid & 31) == 0) wsum[tid >> 5] = lsum;
  __syncthreads();
  if (tid == 0) {
    float t = 0.0f;
    #pragma unroll
    for (int i = 0; i < 8; ++i) t += wsum[i];
    atomicAdd(ws + b, t);
  }
}

// ---------------- Kernel 3: normalize + batch mean ----------------
__global__ void finalize_kernel(const float* __restrict__ ws, float* __restrict__ out) {
  if (threadIdx.x == 0) {
    const unsigned* np = (const unsigned*)ws + 16;
    float cs = 0.0f, rs = 0.0f;
    for (int b = 0; b < B_IMG; ++b) {
      unsigned n = np[b];
      cs += ws[b] / fmaxf((float)n, 1.0f);
      rs += (n > 0u) ? (ws[8 + b] / (4.0f * (float)n)) : 0.0f;
    }
    out[0] = cs * (1.0f / B_IMG);
    out[1] = rs * (1.0f / B_IMG);
  }
}

extern "C" void kernel_launch(void* const* d_in, const int* in_sizes, int n_in,
                              void* d_out, int out_size, void* d_ws, size_t ws_size,
                              hipStream_t stream) {
  (void)in_sizes; (void)n_in; (void)out_size; (void)ws_size;
  const float* cls   = (const float*)d_in[0];  // [B,A,C]
  const float* reg   = (const float*)d_in[1];  // [B,A,4]
  const float* anch  = (const float*)d_in[2];  // [1,A,4]
  const float* annot = (const float*)d_in[3];  // [B,M,5]
  float* ws  = (float*)d_ws;
  float* out = (float*)d_out;

  hipLaunchKernelGGL(init_ws_kernel, dim3(1), dim3(32), 0, stream, ws);
  hipLaunchKernelGGL(assign_kernel, dim3((A_ANCH + 255) / 256, B_IMG), dim3(256), 0, stream,
                     anch, annot, reg, ws);
  // 9.6M floats/image = 1875 blocks * 5 stages * 1024 floats, exact
  hipLaunchKernelGGL(focal_kernel, dim3(1875, B_IMG), dim3(256), 0, stream, cls, ws);
  hipLaunchKernelGGL(finalize_kernel, dim3(1), dim3(32), 0, stream, ws, out);
}